// RelAttGCN_44487271252607
// MI455X (gfx1250) — compile-verified
//
#include <hip/hip_runtime.h>
#include <math.h>

// RelAttGCN on MI455X (gfx1250, wave32).
//
//   out = elu( (exp(A) @ (x*W)) / (exp(A) @ 1) ),  A: COO [2,E], x: [N,128]
//
//  * W diagonal -> folded into finalize:  A@(x*W) = (A@x)*W.
//  * x (51MB) + accumulator (51MB) fit the 192MB L2 -> gather & scatter-add
//    stay L2-resident; HBM cost is one ~120MB stream.
//  * Edge metadata streamed via CDNA5 async-to-LDS (ASYNCcnt, double buffer),
//    consumed as 32-edge groups: 3x ds_load + 1x v_exp per 32 edges, then
//    v_readlane broadcasts (d,s,v) to SGPRs per edge.
//  * hidden accumulates via explicit global_atomic_add_f32 scope:SCOPE_DEV
//    (device-scope RMW executes in L2; no CAS loop, no return traffic).

#define D_DIM   128
#define D_VEC4  (D_DIM / 4)       // 32 float4 per row == one wave32
#define TILE    1024              // edges staged per block per stage
#define WPB     8                 // waves per 256-thread block
#define EDGE_BLOCKS 512

// ---- CDNA5 async copy: 16B per lane, global -> LDS, tracked by ASYNCcnt ----
__device__ __forceinline__ void async_copy_b128(unsigned int lds_off,
                                                const void* gptr) {
    asm volatile("global_load_async_to_lds_b128 %0, %1, off"
                 :: "v"(lds_off),
                    "v"((unsigned long long)(uintptr_t)gptr)
                 : "memory");
}
#define ASYNC_WAIT_3() asm volatile("s_wait_asynccnt 3" ::: "memory")
#define ASYNC_WAIT_0() asm volatile("s_wait_asynccnt 0" ::: "memory")

// ---- device-scope FP32 atomic add (no-return): RMW executes in L2 ----------
__device__ __forceinline__ void gatomic_add_f32(float* p, float v) {
    asm volatile("global_atomic_add_f32 %0, %1, off scope:SCOPE_DEV"
                 :: "v"((unsigned long long)(uintptr_t)p), "v"(v)
                 : "memory");
}

// ------------------------------------------------------------- edge scatter
__global__ __launch_bounds__(256)
void relatt_edge_kernel(const float* __restrict__ x,     // [N, 128]
                        const int*   __restrict__ dst,   // [E]
                        const int*   __restrict__ src,   // [E]
                        const float* __restrict__ vals,  // [E]
                        float*       __restrict__ acc,   // [N, 128] (in d_out)
                        float*       __restrict__ rowsum,// [N]
                        int E) {
    __shared__ __align__(16) int   sD[2][TILE];
    __shared__ __align__(16) int   sS[2][TILE];
    __shared__ __align__(16) float sV[2][TILE];

    const int tid  = threadIdx.x;
    const int lane = tid & 31;
    const int wid  = tid >> 5;
    const int ntiles = E / TILE;

    int t = blockIdx.x;
    int stage = 0;
    if (t < ntiles) {                       // prologue: stage tile 0
        const int o = tid * 4;
        const int b = t * TILE + o;
        async_copy_b128((unsigned int)(uintptr_t)&sD[0][o], dst  + b);
        async_copy_b128((unsigned int)(uintptr_t)&sS[0][o], src  + b);
        async_copy_b128((unsigned int)(uintptr_t)&sV[0][o], vals + b);
    }
    for (; t < ntiles; t += gridDim.x, ++stage) {
        const int buf = stage & 1;
        const int tn  = t + gridDim.x;
        if (tn < ntiles) {                  // stage next tile into other buffer
            const int o  = tid * 4;
            const int b  = tn * TILE + o;
            const int nb = buf ^ 1;
            async_copy_b128((unsigned int)(uintptr_t)&sD[nb][o], dst  + b);
            async_copy_b128((unsigned int)(uintptr_t)&sS[nb][o], src  + b);
            async_copy_b128((unsigned int)(uintptr_t)&sV[nb][o], vals + b);
            ASYNC_WAIT_3();                 // this wave's tile-t copies landed
        } else {
            ASYNC_WAIT_0();
        }
        __syncthreads();                    // tile t visible to all waves

        // 8 waves x 4 groups x 32 edges = 1024 edges per tile
        for (int g = 0; g < TILE / (WPB * 32); ++g) {
            const int ei = wid * (TILE / WPB) + g * 32 + lane;
            const int   dv = sD[buf][ei];
            const int   sv = sS[buf][ei];
            const float vv = expf(sV[buf][ei]);      // 1 v_exp per 32 edges

            // edge-softmax denominator: one scatter-atomic instr per group
            gatomic_add_f32(rowsum + dv, vv);

            for (int k = 0; k < 32; ++k) {           // broadcast edge k
                const int   d = __builtin_amdgcn_readlane(dv, k);
                const int   s = __builtin_amdgcn_readlane(sv, k);
                const float v = __int_as_float(
                    __builtin_amdgcn_readlane(__float_as_int(vv), k));
                const float4 xr =
                    reinterpret_cast<const float4*>(x + (size_t)s * D_DIM)[lane];
                float* p = acc + (size_t)d * D_DIM + lane * 4;
                gatomic_add_f32(p + 0, xr.x * v);
                gatomic_add_f32(p + 1, xr.y * v);
                gatomic_add_f32(p + 2, xr.z * v);
                gatomic_add_f32(p + 3, xr.w * v);
            }
        }
        __syncthreads();                    // all reads of buf done
    }

    // ----- remainder edges (< TILE): plain scalar-load path -----
    const int gw   = __builtin_amdgcn_readfirstlane(blockIdx.x * WPB + wid);
    const int totw = gridDim.x * WPB;
    for (int e = ntiles * TILE + gw; e < E; e += totw) {
        const int   d = dst[e];
        const int   s = src[e];
        const float v = expf(vals[e]);
        const float4 xr =
            reinterpret_cast<const float4*>(x + (size_t)s * D_DIM)[lane];
        float* p = acc + (size_t)d * D_DIM + lane * 4;
        gatomic_add_f32(p + 0, xr.x * v);
        gatomic_add_f32(p + 1, xr.y * v);
        gatomic_add_f32(p + 2, xr.z * v);
        gatomic_add_f32(p + 3, xr.w * v);
        if (lane == 0) gatomic_add_f32(rowsum + d, v);
    }
}

// ---------------------------------------------------------------- zero init
__global__ void relatt_zero_kernel(float* __restrict__ acc,
                                   float* __restrict__ rowsum,
                                   int n_acc4, int n_row) {
    const int tid    = blockIdx.x * blockDim.x + threadIdx.x;
    const int stride = gridDim.x * blockDim.x;
    const float4 z = make_float4(0.f, 0.f, 0.f, 0.f);
    for (int j = tid; j < n_acc4; j += stride)
        reinterpret_cast<float4*>(acc)[j] = z;
    for (int j = tid; j < n_row; j += stride)
        rowsum[j] = 0.f;
}

// ---------------------------------------------------------------- finalize
__global__ void relatt_finalize_kernel(float*       __restrict__ out,
                                       const float* __restrict__ rowsum,
                                       const float* __restrict__ W,
                                       int N) {
    const int tid    = blockIdx.x * blockDim.x + threadIdx.x;
    const int stride = gridDim.x * blockDim.x;
    const int total  = N * D_VEC4;
    for (int j = tid; j < total; j += stride) {
        const int n  = j >> 5;
        const int c4 = j & 31;
        const float inv = 1.0f / rowsum[n];
        float4 a = reinterpret_cast<float4*>(out)[j];
        const float4 w = reinterpret_cast<const float4*>(W)[c4];
        float4 h;
        h.x = a.x * w.x * inv;
        h.y = a.y * w.y * inv;
        h.z = a.z * w.z * inv;
        h.w = a.w * w.w * inv;
        h.x = h.x > 0.f ? h.x : expm1f(h.x);
        h.y = h.y > 0.f ? h.y : expm1f(h.y);
        h.z = h.z > 0.f ? h.z : expm1f(h.z);
        h.w = h.w > 0.f ? h.w : expm1f(h.w);
        reinterpret_cast<float4*>(out)[j] = h;
    }
}

// ------------------------------------------------------------------ launch
extern "C" void kernel_launch(void* const* d_in, const int* in_sizes, int n_in,
                              void* d_out, int out_size, void* d_ws, size_t ws_size,
                              hipStream_t stream) {
    const float* x     = (const float*)d_in[0];          // [N,128]
    const int*   eidx  = (const int*)  d_in[1];          // [2,E] row-major
    const float* evals = (const float*)d_in[2];          // [E]
    const float* W     = (const float*)d_in[3];          // [128]
    float*       out   = (float*)d_out;

    const int D = in_sizes[3];        // 128
    const int N = in_sizes[0] / D;
    const int E = in_sizes[1] / 2;
    const int* dst = eidx;            // row 0
    const int* src = eidx + E;        // row 1

    float* rowsum = (float*)d_ws;     // N floats (400 KB) of scratch

    relatt_zero_kernel<<<4096, 256, 0, stream>>>(out, rowsum, N * D_VEC4, N);
    relatt_edge_kernel<<<EDGE_BLOCKS, 256, 0, stream>>>(x, dst, src, evals,
                                                        out, rowsum, E);
    relatt_finalize_kernel<<<4096, 256, 0, stream>>>(out, rowsum, W, N);
}